// GCNConvLayer_WithEFeat_6605659701696
// MI455X (gfx1250) — compile-verified
//
#include <hip/hip_runtime.h>

typedef float v2f __attribute__((ext_vector_type(2)));
typedef float v8f __attribute__((ext_vector_type(8)));

#define NODE_DIM 128
#define EDGE_DIM 64

// ---------------------------------------------------------------------------
// 1) zero the degree accumulator
__global__ void zero_f32_kernel(float* __restrict__ p, int n) {
    int i = blockIdx.x * blockDim.x + threadIdx.x;
    if (i < n) p[i] = 0.0f;
}

// 2) in-degree count: deg[dst[e]] += 1
__global__ void count_deg_kernel(const int* __restrict__ dst,
                                 float* __restrict__ deg, int E) {
    int e = blockIdx.x * blockDim.x + threadIdx.x;
    if (e < E) atomicAdd(&deg[dst[e]], 1.0f);
}

// 3) norm = (deg + 1)^-0.5
__global__ void norm_kernel(const float* __restrict__ deg,
                            float* __restrict__ nrm, int N) {
    int i = blockIdx.x * blockDim.x + threadIdx.x;
    if (i < N) nrm[i] = 1.0f / sqrtf(deg[i] + 1.0f);
}

// ---------------------------------------------------------------------------
// 4) h = nfeat @ Wn + bn  via V_WMMA_F32_16X16X4_F32, fused with the
//    residual epilogue: out = (h + resid) / (deg + 1).
//    block = 256 threads = 8 wave32; block owns 16 rows x 128 cols;
//    wave w owns the 16x16 tile at cols [16w, 16w+16).
__global__ __launch_bounds__(256) void h_gemm_kernel(
    const float* __restrict__ nfeat, const float* __restrict__ Wn,
    const float* __restrict__ bn, const float* __restrict__ resid,
    const float* __restrict__ deg, float* __restrict__ h,
    float* __restrict__ out, int N) {
    const int r0   = blockIdx.x * 16;
    const int wave = threadIdx.x >> 5;
    const int lane = threadIdx.x & 31;
    const int half = lane >> 4;       // 0: K pair {k0,k0+1}; 1: {k0+2,k0+3}
    const int l16  = lane & 15;       // A: row M; B/C: col N
    const int col0 = wave * 16;
    const int j    = col0 + l16;      // this lane's output column

    const int arow = min(r0 + l16, N - 1);  // clamp keeps EXEC uniform
    const float* __restrict__ aptr = nfeat + (size_t)arow * NODE_DIM;

    const float bias = bn[j];
    v8f c;
#pragma unroll
    for (int i = 0; i < 8; ++i) c[i] = bias;

#pragma unroll 8
    for (int k0 = 0; k0 < NODE_DIM; k0 += 4) {
        const int ka = k0 + 2 * half;
        v2f a;
        a.x = aptr[ka];
        a.y = aptr[ka + 1];
        v2f b;
        b.x = Wn[(size_t)ka * NODE_DIM + j];
        b.y = Wn[(size_t)(ka + 1) * NODE_DIM + j];
        c = __builtin_amdgcn_wmma_f32_16x16x4_f32(
                false, a, false, b, (short)0, c, false, false);
    }

    const float rj = resid[j];

    // D layout: VGPR v -> row M = v + 8*half, col = j
    if (r0 + 16 <= N) {
        // full tile: branch-free epilogue
#pragma unroll
        for (int v = 0; v < 8; ++v) {
            const int row  = r0 + half * 8 + v;
            const float hv = c[v];
            h[(size_t)row * NODE_DIM + j] = hv;
            out[(size_t)row * NODE_DIM + j] = (hv + rj) / (deg[row] + 1.0f);
        }
    } else {
#pragma unroll
        for (int v = 0; v < 8; ++v) {
            const int row = r0 + half * 8 + v;
            if (row < N) {
                const float hv = c[v];
                h[(size_t)row * NODE_DIM + j] = hv;
                out[(size_t)row * NODE_DIM + j] = (hv + rj) / (deg[row] + 1.0f);
            }
        }
    }
}

// ---------------------------------------------------------------------------
// 5) Fused: e-tile = efeat[16 edges] @ We + be  (WMMA, K=64),
//    then msg = nrm[src]*nrm[dst] * (h[src] + e) scatter-added into out[dst].
//    block = 256 threads = 8 waves; block owns 16 edges x 128 cols.
__global__ __launch_bounds__(256) void edge_fused_kernel(
    const float* __restrict__ efeat, const float* __restrict__ We,
    const float* __restrict__ be, const float* __restrict__ h,
    const int* __restrict__ src, const int* __restrict__ dst,
    const float* __restrict__ nrm, float* __restrict__ out, int E) {
    const int e0   = blockIdx.x * 16;
    const int wave = threadIdx.x >> 5;
    const int lane = threadIdx.x & 31;
    const int half = lane >> 4;
    const int l16  = lane & 15;
    const int col0 = wave * 16;
    const int j    = col0 + l16;

    __shared__ int   s_src[16];
    __shared__ int   s_dst[16];
    __shared__ float s_scl[16];
    if (threadIdx.x < 16) {
        const int e = min(e0 + (int)threadIdx.x, E - 1);
        const int s = src[e];
        const int d = dst[e];
        s_src[threadIdx.x] = s;
        s_dst[threadIdx.x] = d;
        s_scl[threadIdx.x] = nrm[s] * nrm[d];
    }
    __syncthreads();

    const int arow = min(e0 + l16, E - 1);
    const float* __restrict__ aptr = efeat + (size_t)arow * EDGE_DIM;

    const float bias = be[j];
    v8f c;
#pragma unroll
    for (int i = 0; i < 8; ++i) c[i] = bias;

#pragma unroll
    for (int k0 = 0; k0 < EDGE_DIM; k0 += 4) {
        const int ka = k0 + 2 * half;
        v2f a;
        a.x = aptr[ka];
        a.y = aptr[ka + 1];
        v2f b;
        b.x = We[(size_t)ka * NODE_DIM + j];
        b.y = We[(size_t)(ka + 1) * NODE_DIM + j];
        c = __builtin_amdgcn_wmma_f32_16x16x4_f32(
                false, a, false, b, (short)0, c, false, false);
    }

    // lane holds e-values for edge m = v + 8*half at feature j
    if (e0 + 16 <= E) {
        // full tile: branch-free gather + scatter
#pragma unroll
        for (int v = 0; v < 8; ++v) {
            const int m = half * 8 + v;
            const int s = s_src[m];
            const int d = s_dst[m];
            const float msg = s_scl[m] * (h[(size_t)s * NODE_DIM + j] + c[v]);
            atomicAdd(&out[(size_t)d * NODE_DIM + j], msg);
        }
    } else {
#pragma unroll
        for (int v = 0; v < 8; ++v) {
            const int m = half * 8 + v;
            if (e0 + m < E) {
                const int s = s_src[m];
                const int d = s_dst[m];
                const float msg = s_scl[m] * (h[(size_t)s * NODE_DIM + j] + c[v]);
                atomicAdd(&out[(size_t)d * NODE_DIM + j], msg);
            }
        }
    }
}

// ---------------------------------------------------------------------------
extern "C" void kernel_launch(void* const* d_in, const int* in_sizes, int n_in,
                              void* d_out, int out_size, void* d_ws, size_t ws_size,
                              hipStream_t stream) {
    const int*   src   = (const int*)d_in[0];
    const int*   dst   = (const int*)d_in[1];
    const float* nfeat = (const float*)d_in[2];
    const float* efeat = (const float*)d_in[3];
    const float* Wn    = (const float*)d_in[4];
    const float* bn    = (const float*)d_in[5];
    const float* We    = (const float*)d_in[6];
    const float* be    = (const float*)d_in[7];
    const float* resid = (const float*)d_in[8];

    const int E = in_sizes[0];
    const int N = in_sizes[2] / NODE_DIM;

    // workspace layout: deg[N] | norm[N] | h[N*128]
    float* ws  = (float*)d_ws;
    float* deg = ws;
    float* nrm = ws + (size_t)N;
    float* h   = ws + (size_t)2 * N;
    float* out = (float*)d_out;

    const int T = 256;

    zero_f32_kernel<<<(N + T - 1) / T, T, 0, stream>>>(deg, N);
    count_deg_kernel<<<(E + T - 1) / T, T, 0, stream>>>(dst, deg, E);
    norm_kernel<<<(N + T - 1) / T, T, 0, stream>>>(deg, nrm, N);

    h_gemm_kernel<<<(N + 15) / 16, T, 0, stream>>>(
        nfeat, Wn, bn, resid, deg, h, out, N);

    edge_fused_kernel<<<(E + 15) / 16, T, 0, stream>>>(
        efeat, We, be, h, src, dst, nrm, out, E);
}